// MoNet_37529424232707
// MI455X (gfx1250) — compile-verified
//
#include <hip/hip_runtime.h>

#define N_NODES 50000
#define N_EDGES 800000
#define IN_F    128
#define HID     64
#define OUT_F   64
#define NCOLS   192   // K * HID == K * OUT_F == 3*64

typedef __attribute__((ext_vector_type(2))) float v2f;
typedef __attribute__((ext_vector_type(8))) float v8f;

// ---------------------------------------------------------------------------
// P[M x 192] = A[M x Kdim] @ W[192 x Kdim]^T  via V_WMMA_F32_16X16X4_F32.
// One wave32 owns a 16-row x 192-col strip: 12 accumulator tiles (v8f each),
// A loaded once per k-step and reused across all 12 N-tiles.
// A layout (16x4 f32, ISA 7.12.2): lanes 0-15 hold K=k+0,k+1; lanes 16-31 hold
// K=k+2,k+3; row = lane%16.  B (4x16) mirrors it.  D: vgpr v -> row v+8*half,
// col = lane%16.
// ---------------------------------------------------------------------------
__global__ void __launch_bounds__(128)
gemm_wmma_f32(const float* __restrict__ A, const float* __restrict__ W,
              float* __restrict__ P, int M, int Kdim)
{
    const int lane = threadIdx.x & 31;
    const int wave = threadIdx.x >> 5;
    const int tile = blockIdx.x * 4 + wave;          // wave-uniform
    if (tile * 16 >= M) return;                      // whole wave exits: EXEC stays all-1s
    const int rowb = tile * 16;
    const int l16  = lane & 15;
    const int half = lane >> 4;

    v8f zero = {0.f,0.f,0.f,0.f,0.f,0.f,0.f,0.f};
    v8f acc[12];
    #pragma unroll
    for (int n = 0; n < 12; ++n) acc[n] = zero;

    const float* arow = A + (size_t)(rowb + l16) * Kdim + 2 * half;
    for (int k = 0; k < Kdim; k += 4) {
        v2f a = *(const v2f*)(arow + k);             // 8B aligned: Kdim%4==0, 2*half even
        #pragma unroll
        for (int n = 0; n < 12; ++n) {
            v2f b = *(const v2f*)(W + (size_t)(n * 16 + l16) * Kdim + k + 2 * half);
            acc[n] = __builtin_amdgcn_wmma_f32_16x16x4_f32(
                false, a, false, b, (short)0, acc[n], false, false);
        }
    }

    #pragma unroll
    for (int n = 0; n < 12; ++n) {
        #pragma unroll
        for (int v = 0; v < 8; ++v) {
            P[(size_t)(rowb + v + 8 * half) * NCOLS + n * 16 + l16] = acc[n][v];
        }
    }
}

// ---------------------------------------------------------------------------
// gw[e,k] = exp(-0.5 * || (tanh(pseudo[e] @ ppw^T + ppb) - mu[k]) * isig[k] ||^2)
// One thread per edge: transcendentals computed once, not 64x redundantly.
// ---------------------------------------------------------------------------
__global__ void __launch_bounds__(256)
compute_gw(const float* __restrict__ pseudo,
           const float* __restrict__ ppw, const float* __restrict__ ppb,
           const float* __restrict__ mu,  const float* __restrict__ isig,
           float* __restrict__ gw, int E)
{
    int e = blockIdx.x * blockDim.x + threadIdx.x;
    if (e >= E) return;
    float u0 = pseudo[2 * e + 0], u1 = pseudo[2 * e + 1];
    float p0 = tanhf(u0 * ppw[0] + u1 * ppw[1] + ppb[0]);
    float p1 = tanhf(u0 * ppw[2] + u1 * ppw[3] + ppb[1]);
    #pragma unroll
    for (int k = 0; k < 3; ++k) {
        float d0 = (p0 - mu[2 * k + 0]) * isig[2 * k + 0];
        float d1 = (p1 - mu[2 * k + 1]) * isig[2 * k + 1];
        gw[3 * e + k] = __expf(-0.5f * (d0 * d0 + d1 * d1));
    }
}

// out[n, o] = bias[o]   (accumulator seed; also clears poisoned buffers)
__global__ void __launch_bounds__(256)
init_bias(float* __restrict__ out, const float* __restrict__ bias, int n_total)
{
    int i = blockIdx.x * blockDim.x + threadIdx.x;
    if (i < n_total) out[i] = bias[i & 63];
}

// ---------------------------------------------------------------------------
// 64 threads per edge: m = sum_k gw[e,k] * proj[src[e], k, o]; atomic scatter
// into agg[dst[e], o].  Gathers are 128B-coalesced per wave and L2-resident
// (proj = 38.4 MB << 192 MB L2).
// ---------------------------------------------------------------------------
__global__ void __launch_bounds__(256)
edge_agg(const float* __restrict__ proj, const float* __restrict__ gw,
         const int* __restrict__ src, const int* __restrict__ dst,
         float* __restrict__ agg, int E)
{
    int t = blockIdx.x * 256 + threadIdx.x;
    int e = t >> 6;
    int o = t & 63;
    if (e >= E) return;
    int s = src[e], d = dst[e];
    float g0 = gw[3 * e + 0], g1 = gw[3 * e + 1], g2 = gw[3 * e + 2];
    const float* pr = proj + (size_t)s * NCOLS;
    float m = g0 * pr[o] + g1 * pr[64 + o] + g2 * pr[128 + o];
    unsafeAtomicAdd(agg + (size_t)d * 64 + o, m);   // -> global_atomic_add_f32
}

// ---------------------------------------------------------------------------
extern "C" void kernel_launch(void* const* d_in, const int* in_sizes, int n_in,
                              void* d_out, int out_size, void* d_ws, size_t ws_size,
                              hipStream_t stream)
{
    const float* feat   = (const float*)d_in[0];
    const float* pseudo = (const float*)d_in[1];
    const int*   eidx   = (const int*)  d_in[2];
    const float* fc_w0  = (const float*)d_in[3];
    const float* bias0  = (const float*)d_in[4];
    const float* mu0    = (const float*)d_in[5];
    const float* isig0  = (const float*)d_in[6];
    const float* ppw0   = (const float*)d_in[7];
    const float* ppb0   = (const float*)d_in[8];
    const float* fc_w1  = (const float*)d_in[9];
    const float* bias1  = (const float*)d_in[10];
    const float* mu1    = (const float*)d_in[11];
    const float* isig1  = (const float*)d_in[12];
    const float* ppw1   = (const float*)d_in[13];
    const float* ppb1   = (const float*)d_in[14];

    const int* src = eidx;
    const int* dst = eidx + N_EDGES;

    // workspace carve-out: proj (38.4MB, reused both layers) | h (12.8MB) | gw (9.6MB)
    float* proj = (float*)d_ws;
    float* h    = proj + (size_t)N_NODES * NCOLS;
    float* gw   = h    + (size_t)N_NODES * HID;
    float* out  = (float*)d_out;

    const int gemm_blocks = (N_NODES / 16 + 3) / 4;          // 3125 tiles, 4 waves/block
    const int gw_blocks   = (N_EDGES + 255) / 256;
    const int ea_blocks   = (N_EDGES * 64) / 256;            // exact: E % 4 == 0
    const int ib0_blocks  = (N_NODES * HID   + 255) / 256;
    const int ib1_blocks  = (N_NODES * OUT_F + 255) / 256;

    // ---- layer 0: GMMConv(IN_F -> HID) ----
    compute_gw<<<gw_blocks, 256, 0, stream>>>(pseudo, ppw0, ppb0, mu0, isig0, gw, N_EDGES);
    gemm_wmma_f32<<<gemm_blocks, 128, 0, stream>>>(feat, fc_w0, proj, N_NODES, IN_F);
    init_bias<<<ib0_blocks, 256, 0, stream>>>(h, bias0, N_NODES * HID);
    edge_agg<<<ea_blocks, 256, 0, stream>>>(proj, gw, src, dst, h, N_EDGES);

    // ---- layer 1: GMMConv(HID -> OUT_F) ----
    compute_gw<<<gw_blocks, 256, 0, stream>>>(pseudo, ppw1, ppb1, mu1, isig1, gw, N_EDGES);
    gemm_wmma_f32<<<gemm_blocks, 128, 0, stream>>>(h, fc_w1, proj, N_NODES, HID);
    init_bias<<<ib1_blocks, 256, 0, stream>>>(out, bias1, N_NODES * OUT_F);
    edge_agg<<<ea_blocks, 256, 0, stream>>>(proj, gw, src, dst, out, N_EDGES);
}